// SPAGCN_46634754900398
// MI455X (gfx1250) — compile-verified
//
#include <hip/hip_runtime.h>

typedef __attribute__((ext_vector_type(2))) float v2f;
typedef __attribute__((ext_vector_type(8))) float v8f;

#define N_NODES 100000
#define DIN 128
#define DH  128
#define DOUT 64
#define K_CL 10
#define KD  128   // contraction dim for both GEMMs

// Hardware f32 atomic add (global_atomic_add_f32, no return).
__device__ __forceinline__ void atomic_add_f32(float* p, float v) {
    unsafeAtomicAdd(p, v);
}

// ---------------------------------------------------------------------------
// Zero-fill (float4 wide)
// ---------------------------------------------------------------------------
__global__ void spagcn_zero4(float4* __restrict__ p, int n4) {
    int i = blockIdx.x * blockDim.x + threadIdx.x;
    int stride = gridDim.x * blockDim.x;
    for (; i < n4; i += stride) p[i] = make_float4(0.f, 0.f, 0.f, 0.f);
}

// ---------------------------------------------------------------------------
// C[16x16 tile] = (RELU?)(A) @ W + bias   via V_WMMA_F32_16X16X4_F32.
// One wave per 16x16 output tile; blockDim.x/32 waves = NDIM/16 col tiles,
// blockIdx.x = 16-row tile. K=128 fully unrolled (32 WMMA). NDIM compile-time
// so every A/B/C access is base + immediate offset (no per-load addr math).
// A 16x4 f32: lanes 0-15 hold (K=k,k+1); lanes 16-31 hold (K=k+2,k+3).
// B 4x16: same K split, N = lane&15. C: VGPR i -> M=i (+8 if lane>=16).
// ---------------------------------------------------------------------------
template<int NDIM, bool RELU>
__global__ void spagcn_wmma_gemm_bias(const float* __restrict__ A,
                                      const float* __restrict__ W,
                                      const float* __restrict__ bias,
                                      float* __restrict__ C) {
    const int lane  = threadIdx.x & 31;
    const int wave  = threadIdx.x >> 5;
    const int row0  = blockIdx.x * 16;
    const int col0  = wave * 16;
    const int lid   = lane & 15;             // M for A-load, N for B/C
    const int koff  = (lane >> 4) << 1;      // 0 or 2
    const int rbase = (lane >> 4) << 3;      // 0 or 8 (C row offset)

    v8f c;
    const float bv = bias[col0 + lid];
    #pragma unroll
    for (int i = 0; i < 8; ++i) c[i] = bv;

    // A: float2 at element (k + koff); base pre-advanced by koff/2
    const float2* __restrict__ arow2 =
        (const float2*)(A + (size_t)(row0 + lid) * KD) + (koff >> 1);
    // B: elements (k+koff)*NDIM and (k+koff+1)*NDIM at fixed column
    const float* __restrict__ wb = W + (size_t)koff * NDIM + col0 + lid;

    #pragma unroll
    for (int k = 0; k < KD; k += 4) {
        float2 av = arow2[k >> 1];
        if (RELU) { av.x = fmaxf(av.x, 0.f); av.y = fmaxf(av.y, 0.f); }
        v2f a; a[0] = av.x; a[1] = av.y;
        v2f b;
        b[0] = wb[(size_t)k * NDIM];
        b[1] = wb[(size_t)(k + 1) * NDIM];
        c = __builtin_amdgcn_wmma_f32_16x16x4_f32(
                /*neg_a=*/false, a, /*neg_b=*/false, b,
                /*c_mod=*/(short)0, c, /*reuse_a=*/false, /*reuse_b=*/false);
    }

    float* __restrict__ cb = C + (size_t)(row0 + rbase) * NDIM + col0 + lid;
    #pragma unroll
    for (int i = 0; i < 8; ++i) cb[(size_t)i * NDIM] = c[i];
}

// ---------------------------------------------------------------------------
// Edge propagate, D=128: wave processes batches of 32 edges. Each lane does one
// coalesced load of its edge's (src,dst,w); __shfl broadcasts per edge. Per
// edge: lane moves float4 -> 4 hw atomics. Tables are L2-resident (51MB<<192MB).
// ---------------------------------------------------------------------------
__global__ void spagcn_prop128(const float* __restrict__ feat,
                               const long long* __restrict__ src,
                               const long long* __restrict__ dst,
                               const float* __restrict__ w,
                               float* __restrict__ out, int E) {
    const int lane = threadIdx.x & 31;
    const int wid = (blockIdx.x * blockDim.x + threadIdx.x) >> 5;
    const int nw  = (gridDim.x * blockDim.x) >> 5;
    for (int base = wid * 32; base < E; base += nw * 32) {
        const int e = base + lane;
        int s = 0, d = 0; float wt = 0.f;
        if (e < E) { s = (int)src[e]; d = (int)dst[e]; wt = w[e]; }
        const int rem = E - base;
        const int cnt = rem < 32 ? rem : 32;
        for (int i = 0; i < cnt; ++i) {
            const int   si = __shfl(s, i);
            const int   di = __shfl(d, i);
            const float wi = __shfl(wt, i);
            const float4 v = ((const float4*)(feat + (size_t)si * 128))[lane];
            float* o = out + (size_t)di * 128 + lane * 4;
            atomic_add_f32(o + 0, v.x * wi);
            atomic_add_f32(o + 1, v.y * wi);
            atomic_add_f32(o + 2, v.z * wi);
            atomic_add_f32(o + 3, v.w * wi);
        }
    }
}

// Edge propagate, D=64: lane moves float2 -> 2 hw atomics per edge.
__global__ void spagcn_prop64(const float* __restrict__ feat,
                              const long long* __restrict__ src,
                              const long long* __restrict__ dst,
                              const float* __restrict__ w,
                              float* __restrict__ out, int E) {
    const int lane = threadIdx.x & 31;
    const int wid = (blockIdx.x * blockDim.x + threadIdx.x) >> 5;
    const int nw  = (gridDim.x * blockDim.x) >> 5;
    for (int base = wid * 32; base < E; base += nw * 32) {
        const int e = base + lane;
        int s = 0, d = 0; float wt = 0.f;
        if (e < E) { s = (int)src[e]; d = (int)dst[e]; wt = w[e]; }
        const int rem = E - base;
        const int cnt = rem < 32 ? rem : 32;
        for (int i = 0; i < cnt; ++i) {
            const int   si = __shfl(s, i);
            const int   di = __shfl(d, i);
            const float wi = __shfl(wt, i);
            const float2 v = ((const float2*)(feat + (size_t)si * 64))[lane];
            float* o = out + (size_t)di * 64 + lane * 2;
            atomic_add_f32(o + 0, v.x * wi);
            atomic_add_f32(o + 1, v.y * wi);
        }
    }
}

// ---------------------------------------------------------------------------
// Student-t soft assignment. ALPHA=1 -> q = (1/(1+sq+1e-8))^2 / 2, normalized.
// mu (10x64) staged in LDS, accessed as float4 (ds_load_b128).
// ---------------------------------------------------------------------------
__global__ void spagcn_student_t(const float* __restrict__ z,
                                 const float* __restrict__ mu,
                                 float* __restrict__ q) {
    __shared__ float4 smu4[K_CL * DOUT / 4];   // 160 float4
    __shared__ float  smu2[K_CL];
    for (int i = threadIdx.x; i < K_CL * DOUT / 4; i += blockDim.x)
        smu4[i] = ((const float4*)mu)[i];
    __syncthreads();
    if (threadIdx.x < K_CL) {
        float s = 0.f;
        #pragma unroll
        for (int j = 0; j < DOUT / 4; ++j) {
            const float4 m = smu4[threadIdx.x * (DOUT / 4) + j];
            s += m.x * m.x + m.y * m.y + m.z * m.z + m.w * m.w;
        }
        smu2[threadIdx.x] = s;
    }
    __syncthreads();

    const int i = blockIdx.x * blockDim.x + threadIdx.x;
    if (i >= N_NODES) return;

    const float4* __restrict__ zi4 = (const float4*)(z + (size_t)i * DOUT);
    float4 zv[DOUT / 4];
    float zz = 0.f;
    #pragma unroll
    for (int j = 0; j < DOUT / 4; ++j) {
        const float4 v = zi4[j];
        zv[j] = v;
        zz += v.x * v.x + v.y * v.y + v.z * v.z + v.w * v.w;
    }

    float qk[K_CL];
    float qs = 0.f;
    #pragma unroll
    for (int k = 0; k < K_CL; ++k) {
        float dot = 0.f;
        #pragma unroll
        for (int j = 0; j < DOUT / 4; ++j) {
            const float4 m = smu4[k * (DOUT / 4) + j];
            dot = fmaf(zv[j].x, m.x, dot);
            dot = fmaf(zv[j].y, m.y, dot);
            dot = fmaf(zv[j].z, m.z, dot);
            dot = fmaf(zv[j].w, m.w, dot);
        }
        const float sq = zz + smu2[k] - 2.f * dot;
        float t = 1.f / (1.f + sq + 1e-8f);   // ALPHA = 1
        t = t * t * 0.5f;                     // q**(ALPHA+1) / 2
        qk[k] = t;
        qs += t;
    }
    const float inv = 1.f / qs;
    float* qo = q + (size_t)i * K_CL;
    #pragma unroll
    for (int k = 0; k < K_CL; ++k) qo[k] = qk[k] * inv;
}

// ---------------------------------------------------------------------------
extern "C" void kernel_launch(void* const* d_in, const int* in_sizes, int n_in,
                              void* d_out, int out_size, void* d_ws, size_t ws_size,
                              hipStream_t stream) {
    const float*     x  = (const float*)d_in[0];
    const long long* ei = (const long long*)d_in[1];   // int64 [2, E]
    const float*     ew = (const float*)d_in[2];
    const float*     W1 = (const float*)d_in[3];
    const float*     b1 = (const float*)d_in[4];
    const float*     W2 = (const float*)d_in[5];
    const float*     b2 = (const float*)d_in[6];
    const float*     mu = (const float*)d_in[7];
    const int E = in_sizes[2];
    const long long* src = ei;
    const long long* dst = ei + E;

    // Workspace layout: h0 [N,128] | h_agg [N,128]; z0 aliases h0 (h0 dead by then)
    float* h0   = (float*)d_ws;
    float* hagg = h0 + (size_t)N_NODES * DH;
    float* z0   = h0;                              // reuse
    float* zout = (float*)d_out;                   // [N,64] accumulated in-place
    float* qout = zout + (size_t)N_NODES * DOUT;   // [N,10]

    // Zero accumulators (every call: replay-safe)
    {
        int n4 = N_NODES * DH / 4;
        spagcn_zero4<<<(n4 + 255) / 256, 256, 0, stream>>>((float4*)hagg, n4);
        int m4 = N_NODES * DOUT / 4;
        spagcn_zero4<<<(m4 + 255) / 256, 256, 0, stream>>>((float4*)zout, m4);
    }

    // GCN layer 1: h0 = x @ W1 + b1   (8 col tiles -> 256 threads = 8 waves)
    spagcn_wmma_gemm_bias<DH, false><<<N_NODES / 16, 256, 0, stream>>>(x, W1, b1, h0);
    // weighted scatter-add over edges
    spagcn_prop128<<<2048, 256, 0, stream>>>(h0, src, dst, ew, hagg, E);
    // GCN layer 2: z0 = relu(hagg) @ W2 + b2   (4 col tiles -> 128 threads)
    spagcn_wmma_gemm_bias<DOUT, true><<<N_NODES / 16, 128, 0, stream>>>(hagg, W2, b2, z0);
    spagcn_prop64<<<2048, 256, 0, stream>>>(z0, src, dst, ew, zout, E);
    // Student-t soft assignment
    spagcn_student_t<<<(N_NODES + 255) / 256, 256, 0, stream>>>(zout, mu, qout);
}